// GMMConv_64707977282161
// MI455X (gfx1250) — compile-verified
//
#include <hip/hip_runtime.h>
#include <math.h>

typedef __attribute__((ext_vector_type(2))) float v2f;
typedef __attribute__((ext_vector_type(4))) float v4f;
typedef __attribute__((ext_vector_type(8))) float v8f;

// ---------------------------------------------------------------------------
// GEMM: out[M, Ncols] = X[M, KD] @ W[Ncols, KD]^T (+ bias)
// One wave computes a 16x64 output tile with V_WMMA_F32_16X16X4_F32.
//
// K-permutation trick: WMMA accumulates over K in any order as long as A and
// B agree per hardware slot. Each lane loads a b128 (4 floats): half-wave
// kh=0 owns K {k0..k0+3}, kh=1 owns {k0+4..k0+7}. Sub-step 0 feeds .xy,
// sub-step 1 feeds .zw — one 16-byte load covers two WMMA K-steps.
//
// C layout (ISA 7.12.2): VGPR r, lanes 0-15 -> (M=r, N=lane);
// lanes 16-31 -> (M=r+8, N=lane-16).
// ---------------------------------------------------------------------------
template <int KD>
__global__ __launch_bounds__(256) void gemm_nt_wmma_f32(
    const float* __restrict__ X,     // [M, KD] row-major
    const float* __restrict__ W,     // [Ncols, KD] row-major (row j = out col j)
    const float* __restrict__ bias,  // [Ncols] or nullptr
    float* __restrict__ out,         // [M, Ncols]
    int M, int Ncols)
{
    const int lane    = threadIdx.x & 31;
    const int wave    = (int)((blockIdx.x * blockDim.x + threadIdx.x) >> 5);
    const int ngroups = Ncols >> 6;            // 64 output cols per wave
    const int mtiles  = (M + 15) >> 4;
    const int mtile   = wave / ngroups;
    const int n0      = (wave % ngroups) << 6;
    if (mtile >= mtiles) return;               // wave-uniform exit (EXEC stays full)

    const int mr = lane & 15;
    const int kh = lane >> 4;                  // half-wave K offset selector

    int arow = mtile * 16 + mr;
    if (arow >= M) arow = M - 1;               // clamp tail reads (stores guarded)
    const float* __restrict__ xr  = X + (size_t)arow * KD + 4 * kh;
    const float* __restrict__ w0p = W + (size_t)(n0 +  0 + mr) * KD + 4 * kh;
    const float* __restrict__ w1p = W + (size_t)(n0 + 16 + mr) * KD + 4 * kh;
    const float* __restrict__ w2p = W + (size_t)(n0 + 32 + mr) * KD + 4 * kh;
    const float* __restrict__ w3p = W + (size_t)(n0 + 48 + mr) * KD + 4 * kh;

    v8f acc0 = {}, acc1 = {}, acc2 = {}, acc3 = {};

    constexpr int MS = KD / 8;                 // macro-steps (8 K-values each)

    // Prime the pipeline.
    v4f a  = *(const v4f*)(xr);
    v4f b0 = *(const v4f*)(w0p);
    v4f b1 = *(const v4f*)(w1p);
    v4f b2 = *(const v4f*)(w2p);
    v4f b3 = *(const v4f*)(w3p);

    #pragma unroll
    for (int m = 0; m < MS; ++m) {
        v4f an = a, c0 = b0, c1 = b1, c2 = b2, c3 = b3;
        if (m + 1 < MS) {                      // prefetch next macro-step
            const int off = 8 * (m + 1);
            an = *(const v4f*)(xr  + off);
            c0 = *(const v4f*)(w0p + off);
            c1 = *(const v4f*)(w1p + off);
            c2 = *(const v4f*)(w2p + off);
            c3 = *(const v4f*)(w3p + off);
        }

        const v2f axy = __builtin_shufflevector(a, a, 0, 1);
        const v2f azw = __builtin_shufflevector(a, a, 2, 3);

        acc0 = __builtin_amdgcn_wmma_f32_16x16x4_f32(false, axy, false,
                 __builtin_shufflevector(b0, b0, 0, 1), (short)0, acc0, false, false);
        acc1 = __builtin_amdgcn_wmma_f32_16x16x4_f32(false, axy, false,
                 __builtin_shufflevector(b1, b1, 0, 1), (short)0, acc1, false, false);
        acc2 = __builtin_amdgcn_wmma_f32_16x16x4_f32(false, axy, false,
                 __builtin_shufflevector(b2, b2, 0, 1), (short)0, acc2, false, false);
        acc3 = __builtin_amdgcn_wmma_f32_16x16x4_f32(false, axy, false,
                 __builtin_shufflevector(b3, b3, 0, 1), (short)0, acc3, false, false);

        acc0 = __builtin_amdgcn_wmma_f32_16x16x4_f32(false, azw, false,
                 __builtin_shufflevector(b0, b0, 2, 3), (short)0, acc0, false, false);
        acc1 = __builtin_amdgcn_wmma_f32_16x16x4_f32(false, azw, false,
                 __builtin_shufflevector(b1, b1, 2, 3), (short)0, acc1, false, false);
        acc2 = __builtin_amdgcn_wmma_f32_16x16x4_f32(false, azw, false,
                 __builtin_shufflevector(b2, b2, 2, 3), (short)0, acc2, false, false);
        acc3 = __builtin_amdgcn_wmma_f32_16x16x4_f32(false, azw, false,
                 __builtin_shufflevector(b3, b3, 2, 3), (short)0, acc3, false, false);

        a = an; b0 = c0; b1 = c1; b2 = c2; b3 = c3;
    }

    // Store 16x64 tile (+ optional bias).
    const int c0i = n0 + mr;
    float bv0 = 0.f, bv1 = 0.f, bv2 = 0.f, bv3 = 0.f;
    if (bias) {
        bv0 = bias[c0i];      bv1 = bias[c0i + 16];
        bv2 = bias[c0i + 32]; bv3 = bias[c0i + 48];
    }
    #pragma unroll
    for (int r = 0; r < 8; ++r) {
        const int rr = mtile * 16 + r + 8 * kh;
        if (rr < M) {
            float* orow = out + (size_t)rr * Ncols;
            orow[c0i]      = acc0[r] + bv0;
            orow[c0i + 16] = acc1[r] + bv1;
            orow[c0i + 32] = acc2[r] + bv2;
            orow[c0i + 48] = acc3[r] + bv3;
        }
    }
}

// ---------------------------------------------------------------------------
// Edge kernel: one wave per edge. Computes the 4 Gaussian mixture weights
// (wave-uniform scalar work), combines the 4 K-slices of proj[row[e]] with
// the mean(1/K) folded in, and atomically scatters 64 floats to out[col[e]].
// Each lane owns channels {2*lane, 2*lane+1}. proj (51 MB) is L2-resident.
// ---------------------------------------------------------------------------
__global__ __launch_bounds__(256) void gmm_edge_scatter(
    const float* __restrict__ proj,    // [N, 4*64]
    const float* __restrict__ pseudo,  // [E, 3]
    const float* __restrict__ mu,      // [4, 3]
    const float* __restrict__ sigma,   // [4, 3]
    const int*   __restrict__ row,
    const int*   __restrict__ col,
    float* __restrict__ out,           // [N, 64]
    int E)
{
    const int lane = threadIdx.x & 31;
    const int e    = (int)((blockIdx.x * blockDim.x + threadIdx.x) >> 5);
    if (e >= E) return;

    const int src = row[e];
    const int dst = col[e];

    const float p0 = pseudo[(size_t)e * 3 + 0];
    const float p1 = pseudo[(size_t)e * 3 + 1];
    const float p2 = pseudo[(size_t)e * 3 + 2];

    float g[4];
    #pragma unroll
    for (int k = 0; k < 4; ++k) {
        float acc = 0.f;
        #pragma unroll
        for (int d = 0; d < 3; ++d) {
            float s = fabsf(sigma[k * 3 + d]);
            s = fmaxf(s, 1e-6f);
            const float p = (d == 0) ? p0 : (d == 1) ? p1 : p2;
            const float c = (p - mu[k * 3 + d]) / s;
            acc = fmaf(c, c, acc);
        }
        g[k] = __expf(-0.5f * acc) * 0.25f;   // 1/K mean folded in
    }

    const float* __restrict__ pr = proj + (size_t)src * 256 + 2 * lane;
    float mx = 0.f, my = 0.f;
    #pragma unroll
    for (int k = 0; k < 4; ++k) {
        v2f p = *(const v2f*)(pr + k * 64);
        mx = fmaf(p.x, g[k], mx);
        my = fmaf(p.y, g[k], my);
    }

    float* op = out + (size_t)dst * 64 + 2 * lane;
    atomicAdd(op + 0, mx);   // global_atomic_add_f32, L2-resident
    atomicAdd(op + 1, my);
}

// ---------------------------------------------------------------------------
// Launch: proj GEMM (ws) -> root GEMM+bias (d_out init) -> edge scatter.
// Stream order guarantees proj/out are fully written before atomics begin.
// ---------------------------------------------------------------------------
extern "C" void kernel_launch(void* const* d_in, const int* in_sizes, int n_in,
                              void* d_out, int out_size, void* d_ws, size_t ws_size,
                              hipStream_t stream) {
    const float* x      = (const float*)d_in[0];
    const float* pseudo = (const float*)d_in[1];
    const float* W_msg  = (const float*)d_in[2];
    const float* mu     = (const float*)d_in[3];
    const float* sigma  = (const float*)d_in[4];
    const float* W_root = (const float*)d_in[5];
    const float* b_root = (const float*)d_in[6];
    const int*   row    = (const int*)d_in[7];
    const int*   col    = (const int*)d_in[8];

    const int IN_C  = 128;
    const int N     = in_sizes[0] / IN_C;        // 50000
    const int E     = in_sizes[7];               // 800000
    const int KOUT  = in_sizes[2] / IN_C;        // K*OUT_C = 256
    const int OUT_C = in_sizes[5] / IN_C;        // 64

    float* proj = (float*)d_ws;                  // N * 256 floats = 51.2 MB
    float* out  = (float*)d_out;                 // N * 64

    const int mtiles = (N + 15) / 16;

    // 1) proj[N, 256] = x @ W_msg^T
    {
        const int waves  = mtiles * (KOUT / 64);
        const int blocks = (waves * 32 + 255) / 256;
        gemm_nt_wmma_f32<128><<<blocks, 256, 0, stream>>>(
            x, W_msg, nullptr, proj, N, KOUT);
    }
    // 2) out[N, 64] = x @ W_root^T + b_root  (initializes d_out)
    {
        const int waves  = mtiles * (OUT_C / 64);
        const int blocks = (waves * 32 + 255) / 256;
        gemm_nt_wmma_f32<128><<<blocks, 256, 0, stream>>>(
            x, W_root, b_root, out, N, OUT_C);
    }
    // 3) edge gather/weight/scatter-add
    {
        const long long waves = E;
        const int blocks = (int)((waves * 32 + 255) / 256);
        gmm_edge_scatter<<<blocks, 256, 0, stream>>>(
            proj, pseudo, mu, sigma, row, col, out, E);
    }
}